// GCNClassifier_20392504721587
// MI455X (gfx1250) — compile-verified
//
#include <hip/hip_runtime.h>
#include <math.h>

typedef __attribute__((ext_vector_type(2))) float v2f;
typedef __attribute__((ext_vector_type(8))) float v8f;

#define DIN 128
#define DH 64
#define DOUT 40
#define DOUTP 48   // padded column count for GEMM2 (3 tiles of 16)

// ---------------------------------------------------------------- CSR build: histogram
__global__ __launch_bounds__(256) void k_zero(int* __restrict__ cnt, int n) {
  int i = blockIdx.x * 256 + threadIdx.x;
  if (i < n) cnt[i] = 0;
}

__global__ __launch_bounds__(256) void k_count(const int* __restrict__ ei,
                                               int* __restrict__ cnt, int nE) {
  int e = blockIdx.x * 256 + threadIdx.x;
  if (e < nE) atomicAdd(&cnt[ei[nE + e]], 1);      // dst row of edge_index
}

__global__ __launch_bounds__(256) void k_dinv(const int* __restrict__ cnt,
                                              float* __restrict__ dinv, int n) {
  int i = blockIdx.x * 256 + threadIdx.x;
  if (i < n) dinv[i] = rsqrtf((float)cnt[i] + 1.0f);  // +1 self-loop
}

// exclusive scan of cnt -> rowptr (3-kernel: block scan, block-sum scan, add offsets)
__global__ __launch_bounds__(256) void k_scan1(const int* __restrict__ cnt,
                                               int* __restrict__ rowptr,
                                               int* __restrict__ bsum, int n) {
  __shared__ int sd[256];
  int i = blockIdx.x * 256 + threadIdx.x;
  int v = (i < n) ? cnt[i] : 0;
  sd[threadIdx.x] = v;
  __syncthreads();
  for (int off = 1; off < 256; off <<= 1) {        // Hillis-Steele inclusive
    int t = (threadIdx.x >= off) ? sd[threadIdx.x - off] : 0;
    __syncthreads();
    sd[threadIdx.x] += t;
    __syncthreads();
  }
  if (i < n) rowptr[i] = sd[threadIdx.x] - v;      // exclusive within block
  if (threadIdx.x == 255) bsum[blockIdx.x] = sd[255];
}

__global__ void k_scan2(int* __restrict__ bsum, int* __restrict__ rowptr,
                        int nb, int n, int nE) {
  if (blockIdx.x == 0 && threadIdx.x == 0) {
    int run = 0;
    for (int b = 0; b < nb; ++b) { int t = bsum[b]; bsum[b] = run; run += t; }
    rowptr[n] = nE;
  }
}

__global__ __launch_bounds__(256) void k_scan3(int* __restrict__ rowptr,
                                               const int* __restrict__ bsum,
                                               int* __restrict__ cursor, int n) {
  int i = blockIdx.x * 256 + threadIdx.x;
  if (i < n) {
    int r = rowptr[i] + bsum[i >> 8];
    rowptr[i] = r;
    cursor[i] = r;
  }
}

__global__ __launch_bounds__(256) void k_fill(const int* __restrict__ ei,
                                              int* __restrict__ cursor,
                                              int* __restrict__ csr, int nE) {
  int e = blockIdx.x * 256 + threadIdx.x;
  if (e < nE) {
    int s = ei[e], d = ei[nE + e];
    int pos = atomicAdd(&cursor[d], 1);
    csr[pos] = s;                                   // dst-sorted source list
  }
}

// ---------------------------------------------------------------- GEMM1: h1 = x @ W1  (N x 128) @ (128 x 64), fp32 WMMA
// W1 staged in LDS K-pair-interleaved: sW[(k/2)*2*DH + c*2 + (k&1)] = W1[k][c]
// so each B fragment {W[kb][c], W[kb+1][c]} is one aligned ds_load_b64.
__global__ __launch_bounds__(256) void k_gemm1(const float* __restrict__ x,
                                               const float* __restrict__ W1,
                                               float* __restrict__ h1, int n) {
  __shared__ float sW[DIN * DH];                   // 32 KB
  for (int i = threadIdx.x; i < DIN * DH; i += 256) {
    int k = i / DH, c = i % DH;
    sW[(k >> 1) * (2 * DH) + c * 2 + (k & 1)] = W1[i];
  }
  __syncthreads();

  const int lane = threadIdx.x & 31;
  const int wave = threadIdx.x >> 5;
  const int rowBase = (blockIdx.x * 8 + wave) * 16;
  if (rowBase >= n) return;                        // uniform per wave -> EXEC full

  const int m  = lane & 15;
  const int kh = (lane >> 4) << 1;                 // K sub-offset: 0 or 2
  int row = rowBase + m;
  if (row > n - 1) row = n - 1;                    // clamp; edge rows masked at store
  const float* __restrict__ xrow = x + (size_t)row * DIN + kh;

  v8f acc0 = {}, acc1 = {}, acc2 = {}, acc3 = {};
  for (int k = 0; k < DIN; k += 4) {
    const v2f a = *(const v2f*)(xrow + k);         // one global_load_b64
    const float* wp = &sW[((k + kh) >> 1) * (2 * DH)];
    v2f b0 = *(const v2f*)(wp + m * 2);            // one ds_load_b64 each
    v2f b1 = *(const v2f*)(wp + (m + 16) * 2);
    v2f b2 = *(const v2f*)(wp + (m + 32) * 2);
    v2f b3 = *(const v2f*)(wp + (m + 48) * 2);
    acc0 = __builtin_amdgcn_wmma_f32_16x16x4_f32(false, a, false, b0, (short)0, acc0, false, false);
    acc1 = __builtin_amdgcn_wmma_f32_16x16x4_f32(false, a, false, b1, (short)0, acc1, false, false);
    acc2 = __builtin_amdgcn_wmma_f32_16x16x4_f32(false, a, false, b2, (short)0, acc2, false, false);
    acc3 = __builtin_amdgcn_wmma_f32_16x16x4_f32(false, a, false, b3, (short)0, acc3, false, false);
  }

  const int rh = (lane >> 4) << 3;                 // C/D: M = v + 8*(lane>=16)
  #pragma unroll
  for (int v = 0; v < 8; ++v) {
    const int r = rowBase + rh + v;
    if (r < n) {
      float* dst = &h1[(size_t)r * DH];
      dst[m]      = acc0[v];
      dst[m + 16] = acc1[v];
      dst[m + 32] = acc2[v];
      dst[m + 48] = acc3[v];
    }
  }
}

// ---------------------------------------------------------------- layer-1 aggregation (CSR gather, no atomics) + self-loop + bias + ReLU
__global__ __launch_bounds__(256) void k_agg1(const float* __restrict__ h1,
                                              const int* __restrict__ rowptr,
                                              const int* __restrict__ csr,
                                              const float* __restrict__ dinv,
                                              const float* __restrict__ bias,
                                              float* __restrict__ h, int n) {
  int node = blockIdx.x * 8 + (threadIdx.x >> 5); // one wave per node
  if (node >= n) return;
  int lane = threadIdx.x & 31;
  int start = rowptr[node], end = rowptr[node + 1];
  float di = dinv[node];
  const int f = 2 * lane;                          // each lane owns 2 consecutive feats

  v2f acc = {0.0f, 0.0f};
  for (int base = start; base < end; base += 32) {
    int idx = base + lane;
    int sv = 0; float wv = 0.0f;
    if (idx < end) { sv = csr[idx]; wv = dinv[sv]; }
    int c = end - base; if (c > 32) c = 32;
    for (int t = 0; t < c; ++t) {
      int s   = __shfl(sv, t, 32);                 // broadcast neighbor + its dinv
      float w = __shfl(wv, t, 32) * di;
      v2f hv = *(const v2f*)(h1 + (size_t)s * DH + f);  // coalesced b64 row read
      acc.x += hv.x * w;
      acc.y += hv.y * w;
    }
  }
  v2f hs = *(const v2f*)(h1 + (size_t)node * DH + f);
  float sl = di * di;
  v2f res;
  res.x = acc.x + hs.x * sl + bias[f];
  res.y = acc.y + hs.y * sl + bias[f + 1];
  res.x = res.x > 0.0f ? res.x : 0.0f;
  res.y = res.y > 0.0f ? res.y : 0.0f;
  *(v2f*)(h + (size_t)node * DH + f) = res;
}

// ---------------------------------------------------------------- GEMM2: t = h @ W2  (N x 64) @ (64 x 40), fp32 WMMA, W2 zero-padded to 48 cols
__global__ __launch_bounds__(256) void k_gemm2(const float* __restrict__ h,
                                               const float* __restrict__ W2,
                                               float* __restrict__ t, int n) {
  __shared__ float sW[DH * DOUTP];                 // 12 KB, interleaved, cols 40..47 = 0
  for (int i = threadIdx.x; i < DH * DOUTP; i += 256) {
    int k = i / DOUTP, c = i % DOUTP;
    float v = (c < DOUT) ? W2[k * DOUT + c] : 0.0f;
    sW[(k >> 1) * (2 * DOUTP) + c * 2 + (k & 1)] = v;
  }
  __syncthreads();

  const int lane = threadIdx.x & 31;
  const int wave = threadIdx.x >> 5;
  const int rowBase = (blockIdx.x * 8 + wave) * 16;
  if (rowBase >= n) return;

  const int m  = lane & 15;
  const int kh = (lane >> 4) << 1;
  int row = rowBase + m;
  if (row > n - 1) row = n - 1;
  const float* __restrict__ hrow = h + (size_t)row * DH + kh;

  v8f acc0 = {}, acc1 = {}, acc2 = {};
  for (int k = 0; k < DH; k += 4) {
    const v2f a = *(const v2f*)(hrow + k);
    const float* wp = &sW[((k + kh) >> 1) * (2 * DOUTP)];
    v2f b0 = *(const v2f*)(wp + m * 2);
    v2f b1 = *(const v2f*)(wp + (m + 16) * 2);
    v2f b2 = *(const v2f*)(wp + (m + 32) * 2);
    acc0 = __builtin_amdgcn_wmma_f32_16x16x4_f32(false, a, false, b0, (short)0, acc0, false, false);
    acc1 = __builtin_amdgcn_wmma_f32_16x16x4_f32(false, a, false, b1, (short)0, acc1, false, false);
    acc2 = __builtin_amdgcn_wmma_f32_16x16x4_f32(false, a, false, b2, (short)0, acc2, false, false);
  }

  const int rh = (lane >> 4) << 3;
  #pragma unroll
  for (int v = 0; v < 8; ++v) {
    const int r = rowBase + rh + v;
    if (r < n) {
      float* dst = &t[(size_t)r * DOUT];
      dst[m]      = acc0[v];
      dst[m + 16] = acc1[v];
      if (m < 8) dst[m + 32] = acc2[v];
    }
  }
}

// ---------------------------------------------------------------- layer-2 aggregation + bias + log_softmax (wave32 per node, fused)
// lanes 0..19 each own features {2*lane, 2*lane+1} of the 40-wide row.
// t rows over-read by idle lanes stay inside the n*64-float t allocation.
__global__ __launch_bounds__(256) void k_agg2(const float* __restrict__ t,
                                              const int* __restrict__ rowptr,
                                              const int* __restrict__ csr,
                                              const float* __restrict__ dinv,
                                              const float* __restrict__ bias,
                                              float* __restrict__ out, int n) {
  int node = blockIdx.x * 8 + (threadIdx.x >> 5);
  if (node >= n) return;
  int lane = threadIdx.x & 31;
  const bool act = lane < 20;
  const int f = 2 * lane;
  int start = rowptr[node], end = rowptr[node + 1];
  float di = dinv[node];

  v2f acc = {0.0f, 0.0f};
  for (int base = start; base < end; base += 32) {
    int idx = base + lane;
    int sv = 0; float wv = 0.0f;
    if (idx < end) { sv = csr[idx]; wv = dinv[sv]; }
    int c = end - base; if (c > 32) c = 32;
    for (int tt = 0; tt < c; ++tt) {
      int s   = __shfl(sv, tt, 32);
      float w = __shfl(wv, tt, 32) * di;
      v2f tv = *(const v2f*)(t + (size_t)s * DOUT + f);
      acc.x += tv.x * w;
      acc.y += tv.y * w;
    }
  }
  v2f ts = *(const v2f*)(t + (size_t)node * DOUT + f);
  float sl = di * di;
  float x0 = acc.x + ts.x * sl + (act ? bias[f] : 0.0f);
  float x1 = acc.y + ts.y * sl + (act ? bias[f + 1] : 0.0f);

  float mx = act ? fmaxf(x0, x1) : -INFINITY;
  #pragma unroll
  for (int off = 16; off > 0; off >>= 1) mx = fmaxf(mx, __shfl_xor(mx, off, 32));

  float s = act ? (__expf(x0 - mx) + __expf(x1 - mx)) : 0.0f;
  #pragma unroll
  for (int off = 16; off > 0; off >>= 1) s += __shfl_xor(s, off, 32);

  float lse = mx + __logf(s);
  if (act) {
    v2f r = { x0 - lse, x1 - lse };
    *(v2f*)(out + (size_t)node * DOUT + f) = r;
  }
}

// ---------------------------------------------------------------- launch
extern "C" void kernel_launch(void* const* d_in, const int* in_sizes, int n_in,
                              void* d_out, int out_size, void* d_ws, size_t ws_size,
                              hipStream_t stream) {
  const float* x  = (const float*)d_in[0];
  const float* W1 = (const float*)d_in[1];
  const float* b1 = (const float*)d_in[2];
  const float* W2 = (const float*)d_in[3];
  const float* b2 = (const float*)d_in[4];
  const int*   ei = (const int*)d_in[5];
  const int n  = in_sizes[0] / DIN;
  const int nE = in_sizes[5] / 2;
  const int nb = (n + 255) / 256;
  float* out = (float*)d_out;

  // workspace: [h1: 64n f32 (reused as t)][h: 64n f32][dinv: n f32]
  //            [cnt: n i32][rowptr: n+1 i32][cursor: n i32][bsum: nb i32][csr: nE i32]
  float* h1    = (float*)d_ws;
  float* h     = h1 + (size_t)n * DH;
  float* dinv  = h + (size_t)n * DH;
  int* cnt     = (int*)(dinv + n);
  int* rowptr  = cnt + n;
  int* cursor  = rowptr + (n + 1);
  int* bsum    = cursor + n;
  int* csr     = bsum + nb;

  dim3 blk(256);
  k_zero  <<<(n + 255) / 256,  blk, 0, stream>>>(cnt, n);
  k_count <<<(nE + 255) / 256, blk, 0, stream>>>(ei, cnt, nE);
  k_dinv  <<<(n + 255) / 256,  blk, 0, stream>>>(cnt, dinv, n);
  k_scan1 <<<nb,               blk, 0, stream>>>(cnt, rowptr, bsum, n);
  k_scan2 <<<1,                1,   0, stream>>>(bsum, rowptr, nb, n, nE);
  k_scan3 <<<(n + 255) / 256,  blk, 0, stream>>>(rowptr, bsum, cursor, n);
  k_fill  <<<(nE + 255) / 256, blk, 0, stream>>>(ei, cursor, csr, nE);
  k_gemm1 <<<(n + 127) / 128,  blk, 0, stream>>>(x, W1, h1, n);
  k_agg1  <<<(n + 7) / 8,      blk, 0, stream>>>(h1, rowptr, csr, dinv, b1, h, n);
  k_gemm2 <<<(n + 127) / 128,  blk, 0, stream>>>(h, W2, h1, n);     // t -> h1
  k_agg2  <<<(n + 7) / 8,      blk, 0, stream>>>(h1, rowptr, csr, dinv, b2, out, n);
}